// FocalLoss_17042430230717
// MI455X (gfx1250) — compile-verified
//
#include <hip/hip_runtime.h>
#include <hip/hip_bf16.h>

// Problem constants (from reference)
#define B_    8
#define K_    100000
#define NANN  32
#define C_    80
#define KC4   (K_ * C_ / 4)        // float4 elements per image = 2,000,000
#define NBA   ((K_ + 255) / 256)   // assign_kernel blocks per image = 391
#define NBC   1024                 // focal_kernel blocks per image

typedef __attribute__((ext_vector_type(2))) float v2f;
typedef __attribute__((ext_vector_type(8))) float v8f;

// ---------------------------------------------------------------------------
// Exact f32 wave32 sum-reduction using V_WMMA_F32_16X16X4_F32.
// Lane v occupies a unique (M,K) slot of the 16x4 A matrix; B = ones 4x16
// => D[m][n] = row-sum of A. Lane0 (+8 acc regs) holds rows 0..7 at N=0,
// lane16 rows 8..15, so 8 local adds + one lane|16 gather finish the sum.
// Exact f32 arithmetic — same precision as a shuffle tree.
// ---------------------------------------------------------------------------
__device__ __forceinline__ float wave_reduce_sum(float v) {
#if __has_builtin(__builtin_amdgcn_wmma_f32_16x16x4_f32)
    v2f a;  a[0] = v;    a[1] = 0.0f;
    v2f o;  o[0] = 1.0f; o[1] = 1.0f;
    v8f c = {0.f, 0.f, 0.f, 0.f, 0.f, 0.f, 0.f, 0.f};
    c = __builtin_amdgcn_wmma_f32_16x16x4_f32(
            /*neg_a=*/false, a, /*neg_b=*/false, o,
            /*c_mod=*/(short)0, c, /*reuse_a=*/false, /*reuse_b=*/false);
    float s = c[0] + c[1] + c[2] + c[3] + c[4] + c[5] + c[6] + c[7];
    s += __shfl_down(s, 16, 32);       // lane l (<16) += lane l+16
    return s;                          // valid in lanes 0..15
#else
    for (int d = 16; d > 0; d >>= 1) v += __shfl_down(v, d, 32);
    return v;
#endif
}

// Block-level sum; result valid on thread 0. No atomics (deterministic).
__device__ __forceinline__ float block_reduce(float v, float* red) {
    const int lane = threadIdx.x & 31;
    const int wid  = threadIdx.x >> 5;
    float ws = wave_reduce_sum(v);
    if (lane == 0) red[wid] = ws;
    __syncthreads();
    float t = 0.f;
    if (threadIdx.x == 0) {
        const int nw = (blockDim.x + 31) >> 5;
        for (int w = 0; w < nw; ++w) t += red[w];
    }
    __syncthreads();   // allow safe reuse of `red`
    return t;
}

// ---------------------------------------------------------------------------
// Kernel 1: anchor<->GT assignment + regression smooth-L1 + num_pos.
// tcode[b*K+k]: >=0 positive (class label), -2 negative (all-0 targets),
//               -1 ignore (0.4 <= iou_max < 0.5)
// posPart/regPart: one deterministic partial per block.
// ---------------------------------------------------------------------------
__global__ void assign_kernel(const float* __restrict__ anchors,
                              const float* __restrict__ regressions,
                              const float* __restrict__ annotations,
                              int* __restrict__ tcode,
                              float* __restrict__ posPart,
                              float* __restrict__ regPart) {
    __shared__ float ann[NANN * 5];
    __shared__ float red[8];

    const int b = blockIdx.y;
    const int k = blockIdx.x * blockDim.x + threadIdx.x;
    if (threadIdx.x < NANN * 5)
        ann[threadIdx.x] = annotations[b * NANN * 5 + threadIdx.x];
    __syncthreads();

    const bool active = (k < K_);
    const int  kk = active ? k : 0;
    const float ax1 = anchors[kk * 4 + 0], ay1 = anchors[kk * 4 + 1];
    const float ax2 = anchors[kk * 4 + 2], ay2 = anchors[kk * 4 + 3];
    const float aw = ax2 - ax1, ah = ay2 - ay1;
    const float area_a = aw * ah;

    float best = -2.0f; int bestn = 0;
    #pragma unroll 4
    for (int n = 0; n < NANN; ++n) {
        const float bx1 = ann[n * 5 + 0], by1 = ann[n * 5 + 1];
        const float bx2 = ann[n * 5 + 2], by2 = ann[n * 5 + 3];
        const float lb  = ann[n * 5 + 4];
        const float iw = fmaxf(fminf(ax2, bx2) - fmaxf(ax1, bx1), 0.f);
        const float ih = fmaxf(fminf(ay2, by2) - fmaxf(ay1, by1), 0.f);
        const float inter = iw * ih;
        const float area_b = (bx2 - bx1) * (by2 - by1);
        const float un = fmaxf(area_a + area_b - inter, 1e-8f);
        float iou = inter / un;
        if (lb < 0.f) iou = -1.0f;                   // padded GT
        if (iou > best) { best = iou; bestn = n; }   // first-max == argmax
    }

    const bool pos = active && (best >= 0.5f);
    if (active) {
        const int code = pos ? (int)ann[bestn * 5 + 4]
                             : ((best < 0.4f) ? -2 : -1);
        tcode[(size_t)b * K_ + k] = code;
    }

    float regc = 0.f;
    if (pos) {
        const float gx1 = ann[bestn * 5 + 0], gy1 = ann[bestn * 5 + 1];
        const float gx2 = ann[bestn * 5 + 2], gy2 = ann[bestn * 5 + 3];
        const float gw = fmaxf(gx2 - gx1, 1.f), gh = fmaxf(gy2 - gy1, 1.f);
        const float gcx = gx1 + 0.5f * gw, gcy = gy1 + 0.5f * gh;
        const float acx = ax1 + 0.5f * aw, acy = ay1 + 0.5f * ah;
        const float t0 = ((gcx - acx) / aw) / 0.1f;
        const float t1 = ((gcy - acy) / ah) / 0.1f;
        const float t2 = __logf(gw / aw) / 0.2f;
        const float t3 = __logf(gh / ah) / 0.2f;
        const float* r = regressions + ((size_t)b * K_ + k) * 4;
        const float th = 1.0f / 9.0f, cc = 0.5f / 9.0f;
        float d;
        d = fabsf(t0 - r[0]); regc += (d <= th) ? 4.5f * d * d : d - cc;
        d = fabsf(t1 - r[1]); regc += (d <= th) ? 4.5f * d * d : d - cc;
        d = fabsf(t2 - r[2]); regc += (d <= th) ? 4.5f * d * d : d - cc;
        d = fabsf(t3 - r[3]); regc += (d <= th) ? 4.5f * d * d : d - cc;
    }

    const float npb = block_reduce(pos ? 1.f : 0.f, red);
    if (threadIdx.x == 0) posPart[b * NBA + blockIdx.x] = npb;
    const float rgb = block_reduce(regc, red);
    if (threadIdx.x == 0) regPart[b * NBA + blockIdx.x] = rgb;
}

// ---------------------------------------------------------------------------
// Kernel 2: focal classification loss over the 256 MB stream (HBM-bound).
// float4 loads (4 | C=80 -> a float4 never crosses an anchor boundary).
// Negative / non-matched class: 0.75 * sqrt(p) * -log(1-p)
// Matched class:                0.25 * sqrt(1-p) * -log(p)
// Ignore anchors contribute 0.
// ---------------------------------------------------------------------------
__global__ void focal_kernel(const float4* __restrict__ cls,
                             const int* __restrict__ tcode,
                             float* __restrict__ clsPart) {
    __shared__ float red[8];
    const int b = blockIdx.y;
    const float4* base = cls   + (size_t)b * KC4;
    const int*    tc   = tcode + (size_t)b * K_;
    const int stride = gridDim.x * blockDim.x;

    float s = 0.f;
    for (int i = blockIdx.x * blockDim.x + threadIdx.x; i < KC4; i += stride) {
        __builtin_prefetch(base + i + stride, 0, 1);   // global_prefetch_b8
        const float4 v = base[i];
        const int k    = i / 20;            // 20 float4 per anchor (C=80)
        const int c0   = (i % 20) * 4;
        const int code = tc[k];
        if (code != -1) {
            const float pv[4] = {v.x, v.y, v.z, v.w};
            #pragma unroll
            for (int j = 0; j < 4; ++j) {
                const float p = fminf(fmaxf(pv[j], 1e-4f), 1.0f - 1e-4f);
                float e;
                if (code == c0 + j)      // matched class of a positive anchor
                    e = 0.25f * sqrtf(1.0f - p) * (-__logf(p));
                else                     // target 0 (code==-2 or other class)
                    e = 0.75f * sqrtf(p) * (-__logf(1.0f - p));
                s += e;
            }
        }
    }
    const float bs = block_reduce(s, red);
    if (threadIdx.x == 0) clsPart[b * NBC + blockIdx.x] = bs;
}

// ---------------------------------------------------------------------------
// Kernel 3: deterministic finalize. Threads 0..7 serially sum their image's
// partials in fixed order, then thread 0 writes the two batch means.
// ---------------------------------------------------------------------------
__global__ void finalize_kernel(const float* __restrict__ posPart,
                                const float* __restrict__ regPart,
                                const float* __restrict__ clsPart,
                                float* __restrict__ out) {
    __shared__ float cl[B_], rl[B_];
    const int t = threadIdx.x;
    if (t < B_) {
        float np = 0.f, rs = 0.f, cs = 0.f;
        for (int i = 0; i < NBA; ++i) np += posPart[t * NBA + i];
        for (int i = 0; i < NBA; ++i) rs += regPart[t * NBA + i];
        for (int i = 0; i < NBC; ++i) cs += clsPart[t * NBC + i];
        cl[t] = cs / fmaxf(np, 1.f);
        rl[t] = (np > 0.f) ? (rs / fmaxf(np * 4.f, 1.f)) : 0.f;
    }
    __syncthreads();
    if (t == 0) {
        float cm = 0.f, rm = 0.f;
        for (int b = 0; b < B_; ++b) { cm += cl[b]; rm += rl[b]; }
        out[0] = cm / (float)B_;
        out[1] = rm / (float)B_;
    }
}

// ---------------------------------------------------------------------------
extern "C" void kernel_launch(void* const* d_in, const int* in_sizes, int n_in,
                              void* d_out, int out_size, void* d_ws, size_t ws_size,
                              hipStream_t stream) {
    const float* classifications = (const float*)d_in[0];   // [B,K,C]
    const float* regressions     = (const float*)d_in[1];   // [B,K,4]
    const float* anchors         = (const float*)d_in[2];   // [1,K,4]
    const float* annotations     = (const float*)d_in[3];   // [B,32,5]

    // Workspace layout (all slots written every call -> no init needed):
    //   [0, B*K)                 int   tcode
    //   then float posPart[B*NBA], regPart[B*NBA], clsPart[B*NBC]
    int*   tcode   = (int*)d_ws;
    float* posPart = (float*)d_ws + (size_t)B_ * K_;
    float* regPart = posPart + (size_t)B_ * NBA;
    float* clsPart = regPart + (size_t)B_ * NBA;

    dim3 gA(NBA, B_);
    assign_kernel<<<gA, 256, 0, stream>>>(anchors, regressions, annotations,
                                          tcode, posPart, regPart);

    dim3 gB(NBC, B_);
    focal_kernel<<<gB, 256, 0, stream>>>((const float4*)classifications,
                                         tcode, clsPart);

    finalize_kernel<<<1, 32, 0, stream>>>(posPart, regPart, clsPart,
                                          (float*)d_out);
}